// KNN_60026462929129
// MI455X (gfx1250) — compile-verified
//
#include <hip/hip_runtime.h>
#include <math.h>

#define D_EMB 512
#define BLOCK 256
#define WAVES_PER_BLOCK (BLOCK / 32)

typedef float v2f __attribute__((ext_vector_type(2)));
typedef float v8f __attribute__((ext_vector_type(8)));

// ---------------------------------------------------------------------------
// Phase 1: distances via V_WMMA_F32_16X16X4_F32.
// Each wave owns a tile of 16 database rows. Per k-chunk of 4:
//   A[m,k] = X[row0+m, kc+k]   (ISA layout: lane L -> row L%16, k = kc+2*(L/16))
//   B[k,n] = x[kc+k]           (query broadcast across all 16 columns)
// D accumulates dot(row_m, x) in fp32 (identical in every column).
// Row norms accumulate in VALU from the same A values.
// ---------------------------------------------------------------------------
__global__ __launch_bounds__(BLOCK) void knn_dist_kernel(
    const float* __restrict__ Xemb, const float* __restrict__ xq,
    float* __restrict__ dist, int N) {
  __shared__ float xs[D_EMB];
  __shared__ float red[BLOCK];
  const int tid = threadIdx.x;
  for (int i = tid; i < D_EMB; i += BLOCK) xs[i] = xq[i];
  __syncthreads();
  float p = 0.f;
  for (int i = tid; i < D_EMB; i += BLOCK) p += xs[i] * xs[i];
  red[tid] = p;
  __syncthreads();
  for (int s = BLOCK / 2; s > 0; s >>= 1) {
    if (tid < s) red[tid] += red[tid + s];
    __syncthreads();
  }
  const float xn = red[0];  // ||x||^2

  const int lane  = tid & 31;
  const int wave  = blockIdx.x * WAVES_PER_BLOCK + (tid >> 5);
  const int nwave = gridDim.x * WAVES_PER_BLOCK;
  const int tiles = (N + 15) >> 4;
  const int m     = lane & 15;
  const int half  = lane >> 4;

  for (int tile = wave; tile < tiles; tile += nwave) {
    const int row0 = tile << 4;
    if (row0 + 16 <= N) {
      const float* rowp = Xemb + (size_t)(row0 + m) * D_EMB;
      __builtin_prefetch(Xemb + (size_t)(row0 + nwave * 16 + m) * D_EMB, 0, 3);
      v8f c = {};
      float na = 0.f;  // per-lane half-row sum of squares
#pragma unroll 4
      for (int kc = 0; kc < D_EMB; kc += 4) {
        const int koff = kc + 2 * half;
        v2f a, b;
        a.x = rowp[koff];
        a.y = rowp[koff + 1];
        b.x = xs[koff];
        b.y = xs[koff + 1];
        na = fmaf(a.x, a.x, fmaf(a.y, a.y, na));
        c = __builtin_amdgcn_wmma_f32_16x16x4_f32(
            /*neg_a=*/false, a, /*neg_b=*/false, b,
            /*c_mod=*/(short)0, c, /*reuse_a=*/false, /*reuse_b=*/false);
      }
      // Combine the two K-halves of the row norm: lane L gets ||row L%16||^2.
      float nfull = na + __shfl_xor(na, 16, 32);
      // Store lanes: lane r -> row r (norm at lane r),
      //              lane 16+r -> row 8+r (norm lives at lane 8+r).
      float nrm = __shfl(nfull, (lane < 16) ? lane : (lane - 8), 32);
#pragma unroll
      for (int r = 0; r < 8; ++r) {
        if (lane == r || lane == 16 + r) {
          const int row = (lane < 16) ? r : (8 + r);
          // D layout: VGPR r holds rows r (lanes 0-15) / 8+r (lanes 16-31).
          float d2 = nrm - 2.0f * c[r] + xn;
          dist[row0 + row] = sqrtf(fmaxf(d2, 0.0f));
        }
      }
    } else {  // tail tile (never hit for N=200000, kept for safety)
      const int row = row0 + lane;
      if (row < N) {
        const float* rowp = Xemb + (size_t)row * D_EMB;
        float d2 = 0.f;
        for (int kk = 0; kk < D_EMB; ++kk) {
          float df = rowp[kk] - xs[kk];
          d2 = fmaf(df, df, d2);
        }
        dist[row] = sqrtf(d2);
      }
    }
  }
}

// ---------------------------------------------------------------------------
// Phase 2: iterated argmin. Key = (dist_bits << 32) | index.
// dist >= 0 so IEEE bits are monotone; low-bits index reproduces jax's
// lower-index tie-break. atomicMin is order-independent => deterministic.
// ---------------------------------------------------------------------------
__global__ void knn_init_best(unsigned long long* best) { *best = ~0ull; }

__global__ __launch_bounds__(BLOCK) void knn_argmin_kernel(
    const float* __restrict__ dist, int N, unsigned long long* best) {
  __shared__ unsigned long long sd[BLOCK];
  unsigned long long k = ~0ull;
  for (int i = blockIdx.x * blockDim.x + threadIdx.x; i < N;
       i += gridDim.x * blockDim.x) {
    unsigned long long key =
        ((unsigned long long)__float_as_uint(dist[i]) << 32) | (unsigned int)i;
    k = key < k ? key : k;
  }
  sd[threadIdx.x] = k;
  __syncthreads();
  for (int s = BLOCK / 2; s > 0; s >>= 1) {
    if (threadIdx.x < s) {
      unsigned long long o = sd[threadIdx.x + s];
      if (o < sd[threadIdx.x]) sd[threadIdx.x] = o;
    }
    __syncthreads();
  }
  if (threadIdx.x == 0) atomicMin(best, sd[0]);
}

__global__ void knn_select_kernel(float* dist, const int* __restrict__ cls,
                                  float* out_cls, float* out_dist, int* sel,
                                  unsigned long long* best, int j) {
  unsigned long long b = *best;
  unsigned int idx = (unsigned int)(b & 0xffffffffu);
  out_dist[j] = __uint_as_float((unsigned int)(b >> 32));
  out_cls[j] = (float)cls[idx];
  sel[j] = (int)idx;
  dist[idx] = __uint_as_float(0x7f800000u);  // +inf: exclude from next round
  *best = ~0ull;                             // reset for next round
}

__global__ __launch_bounds__(BLOCK) void knn_gather_kernel(
    const float* __restrict__ Xemb, const int* __restrict__ sel,
    float* __restrict__ out_emb) {
  const int j = blockIdx.x;
  const size_t idx = (size_t)sel[j];
  for (int t = threadIdx.x; t < D_EMB; t += blockDim.x)
    out_emb[(size_t)j * D_EMB + t] = Xemb[idx * D_EMB + t];
}

// ---------------------------------------------------------------------------
extern "C" void kernel_launch(void* const* d_in, const int* in_sizes, int n_in,
                              void* d_out, int out_size, void* d_ws,
                              size_t ws_size, hipStream_t stream) {
  const float* xq   = (const float*)d_in[0];
  const float* Xemb = (const float*)d_in[1];
  const int*   Xcls = (const int*)d_in[2];
  const int N = in_sizes[1] / D_EMB;             // 200000
  const int k = out_size / (D_EMB + 2);          // 32 (k*512 + k + k floats)

  float* out_emb  = (float*)d_out;
  float* out_cls  = out_emb + (size_t)k * D_EMB;
  float* out_dist = out_cls + k;

  float* dist = (float*)d_ws;
  size_t off = ((size_t)N * sizeof(float) + 7) & ~(size_t)7;
  unsigned long long* best = (unsigned long long*)((char*)d_ws + off);
  int* sel = (int*)(best + 1);

  const int tiles = (N + 15) / 16;
  const int dblocks = (tiles + WAVES_PER_BLOCK - 1) / WAVES_PER_BLOCK;
  knn_dist_kernel<<<dblocks, BLOCK, 0, stream>>>(Xemb, xq, dist, N);
  knn_init_best<<<1, 1, 0, stream>>>(best);
  for (int j = 0; j < k; ++j) {
    knn_argmin_kernel<<<256, BLOCK, 0, stream>>>(dist, N, best);
    knn_select_kernel<<<1, 1, 0, stream>>>(dist, Xcls, out_cls, out_dist, sel,
                                           best, j);
  }
  knn_gather_kernel<<<k, BLOCK, 0, stream>>>(Xemb, sel, out_emb);
}